// OT_Loss_72791105733258
// MI455X (gfx1250) — compile-verified
//
#include <hip/hip_runtime.h>
#include <math.h>

#define OUTN   128
#define NPTS   1024
#define GRIDN  (OUTN*OUTN)
#define REGC   10.0f
#define ITERS  100
#define MEPS   1e-16f
#define TPB    512
#define NWAVE  16

#define KPAD   68     // staging row stride (k-contiguous), conflict-free b64
#define VPAD   132    // s_V row stride (gx-contiguous), conflict-free b64

typedef float v2f __attribute__((ext_vector_type(2)));
typedef float v8f __attribute__((ext_vector_type(8)));

__device__ __forceinline__ float gridc(int i) { return (float)(i * 4 + 2); }

__device__ __forceinline__ v8f v8zero() {
    v8f z = {0.f,0.f,0.f,0.f,0.f,0.f,0.f,0.f};
    return z;
}

__device__ __forceinline__ v8f wmma_f32(v2f a, v2f b, v8f c) {
    // D = A(16x4 f32) x B(4x16 f32) + C(16x16 f32)
    return __builtin_amdgcn_wmma_f32_16x16x4_f32(false, a, false, b, (short)0, c, false, false);
}

__device__ __forceinline__ v2f ld2(const float* p) { return *(const v2f*)p; }

// ---------------------------------------------------------------------------
// Kernel 1: precompute Kx[b][n][g] = exp(-(x-c_g)^2/REG), Ky likewise with y.
// ---------------------------------------------------------------------------
__global__ void prep_kernel(const float* __restrict__ pts,
                            float* __restrict__ Kx, float* __restrict__ Ky) {
    int idx = blockIdx.x * blockDim.x + threadIdx.x;
    if (idx >= 8 * NPTS * OUTN) return;
    int g  = idx & (OUTN - 1);
    int bn = idx >> 7;               // b*N + n
    float x = pts[bn * 2 + 0];
    float y = pts[bn * 2 + 1];
    float dx = x - gridc(g);
    float dy = y - gridc(g);
    Kx[idx] = expf(-(dx * dx) / REGC);
    Ky[idx] = expf(-(dy * dy) / REGC);
}

// ---------------------------------------------------------------------------
// block-wide reduce with broadcast (valid in all threads)
// ---------------------------------------------------------------------------
__device__ float blk_reduce(float v, float* red, float* bc, int tid) {
    #pragma unroll
    for (int m = 16; m >= 1; m >>= 1) v += __shfl_xor(v, m, 32);
    __syncthreads();
    if ((tid & 31) == 0) red[tid >> 5] = v;
    __syncthreads();
    if (tid == 0) {
        float r = 0.f;
        for (int i = 0; i < NWAVE; ++i) r += red[i];
        bc[0] = r;
    }
    __syncthreads();
    return bc[0];
}

// ---------------------------------------------------------------------------
// Kernel 2: persistent Sinkhorn, one workgroup (16 waves) per image.
// ---------------------------------------------------------------------------
__launch_bounds__(TPB, 1)
__global__ void sinkhorn_kernel(const float* __restrict__ nd,
                                const float* __restrict__ und,
                                const float* __restrict__ pts,
                                const float* __restrict__ Kx,
                                const float* __restrict__ Ky,
                                float* __restrict__ res) {
    __shared__ float s_A[OUTN * KPAD];   // 34KB : staging (u*Ky)[col][k] / W block
    __shared__ float s_B[OUTN * KPAD];   // 34KB : staging Kx[col][k]
    __shared__ float s_V[OUTN * VPAD];   // 66KB : v row-major, s_V[gy*VPAD+gx]
    __shared__ float s_u[NPTS];          // 4KB
    __shared__ float s_x[NPTS];          // 4KB
    __shared__ float s_y[NPTS];          // 4KB
    __shared__ float s_red[NWAVE];
    __shared__ float s_bc[1];

    const int b    = blockIdx.x;
    const int tid  = threadIdx.x;
    const int wv   = tid >> 5;       // wave id 0..15
    const int lane = tid & 31;
    const int lh   = lane >> 4;      // lane half (k-pair select / row+8)
    const int lm   = lane & 15;      // matrix row (A) or column (B/C/D)

    const float* ndb  = nd  + b * GRIDN;
    const float* undb = und + b * GRIDN;
    const float* ptsb = pts + b * NPTS * 2;
    const float* Kxb  = Kx  + b * NPTS * OUTN;
    const float* Kyb  = Ky  + b * NPTS * OUTN;

    for (int n = tid; n < NPTS; n += TPB) {
        s_x[n] = ptsb[2 * n];
        s_y[n] = ptsb[2 * n + 1];
        s_u[n] = 1.0f / (float)NPTS;
    }
    __syncthreads();

    // phase-1 tiles: I = wv>>1 (gy tile), J = (wv&1)*4 + t (gx tile)  -> A shared over t
    const int p1I  = wv >> 1;
    const int p1J0 = (wv & 1) * 4;
    // phase-2 tiles: ntile = wv>>2 (n tile), gy tiles (wv&3)*2 + t    -> A shared over t
    const int ntile = wv >> 2;
    const int gyq   = wv & 3;

    const float* paseA = &s_A[(p1I * 16 + lm) * KPAD + 2 * lh];

    for (int it = 0; it < ITERS; ++it) {
        // ===== phase 1:  T[gy,gx] = sum_n (u*Ky)[n,gy]*Kx[n,gx];  v = nd/(T+eps)
        v8f acc[4];
        #pragma unroll
        for (int t = 0; t < 4; ++t) acc[t] = v8zero();

        for (int c = 0; c < 16; ++c) {           // K=1024 in chunks of 64
            __syncthreads();
            for (int idx = tid; idx < 64 * OUTN; idx += TPB) {
                int k = idx >> 7, col = idx & 127;
                int n = c * 64 + k;
                s_A[col * KPAD + k] = s_u[n] * Kyb[n * OUTN + col];  // (u*Ky)[col][k]
                s_B[col * KPAD + k] = Kxb[n * OUTN + col];           // Kx[col][k]
            }
            __syncthreads();
            const float* pb0 = &s_B[(p1J0 * 16 + lm) * KPAD + 2 * lh];
            #pragma unroll 4
            for (int kk = 0; kk < 16; ++kk) {
                v2f av = ld2(paseA + kk * 4);                 // A frag shared by 4 tiles
                #pragma unroll
                for (int t = 0; t < 4; ++t) {
                    v2f bv = ld2(pb0 + t * (16 * KPAD) + kk * 4);
                    acc[t] = wmma_f32(av, bv, acc[t]);
                }
            }
        }
        __syncthreads();
        #pragma unroll
        for (int t = 0; t < 4; ++t) {
            int gx = (p1J0 + t) * 16 + lm;
            #pragma unroll
            for (int r = 0; r < 8; ++r) {
                int gy = p1I * 16 + r + 8 * lh;
                float T = acc[t][r];
                s_V[gy * VPAD + gx] = ndb[gy * OUTN + gx] / (T + MEPS);
            }
        }
        __syncthreads();

        // ===== phase 2:  s[n] = sum_gy Ky[n,gy]*(sum_gx Kx[n,gx]*V[gy,gx]);  u = a/(s+eps)
        for (int nb = 0; nb < 16; ++nb) {        // 16 blocks of 64 n-rows
            int rbase = nb * 64;
            int row = rbase + ntile * 16 + lm;
            const float* pga = Kxb + row * OUTN + 2 * lh;
            const float* pv0 = &s_V[((gyq * 2) * 16 + lm) * VPAD + 2 * lh];
            v8f a2[2];
            a2[0] = v8zero(); a2[1] = v8zero();
            #pragma unroll 4
            for (int kk = 0; kk < 32; ++kk) {    // K = gx = 128
                v2f av = ld2(pga + kk * 4);                   // Kx row (global b64)
                #pragma unroll
                for (int t = 0; t < 2; ++t) {
                    v2f bv = ld2(pv0 + t * (16 * VPAD) + kk * 4);
                    a2[t] = wmma_f32(av, bv, a2[t]);
                }
            }
            // W tiles -> s_A (reused as plain [64][128])
            #pragma unroll
            for (int t = 0; t < 2; ++t) {
                int gyt = gyq * 2 + t;
                #pragma unroll
                for (int r = 0; r < 8; ++r) {
                    int wr = ntile * 16 + r + 8 * lh;
                    s_A[wr * OUTN + gyt * 16 + lm] = a2[t][r];
                }
            }
            __syncthreads();
            // reduce: s[n] = sum_gy Ky*W  (8 threads per row)
            int rrow = tid >> 3;
            int q = tid & 7;
            float sacc = 0.0f;
            #pragma unroll 4
            for (int j = 0; j < 16; ++j) {
                int gy = q * 16 + j;
                sacc += Kyb[(rbase + rrow) * OUTN + gy] * s_A[rrow * OUTN + gy];
            }
            sacc += __shfl_xor(sacc, 1, 32);
            sacc += __shfl_xor(sacc, 2, 32);
            sacc += __shfl_xor(sacc, 4, 32);
            if (q == 0) s_u[rbase + rrow] = (1.0f / (float)NPTS) / (sacc + MEPS);
            __syncthreads();
        }
    }

    // ===== wd = sum dis*P, P = u Ky Kx v, dis = dy+dx  (separable -> 2 WMMA passes)
    float wdp = 0.0f;
    for (int nb = 0; nb < 16; ++nb) {
        int rbase = nb * 64;
        int row = rbase + ntile * 16 + lm;
        const float* pga = Kxb + row * OUTN + 2 * lh;
        const float* pv0 = &s_V[((gyq * 2) * 16 + lm) * VPAD + 2 * lh];
        float xr = s_x[row];
        v8f qa[2], qb[2];
        qa[0] = v8zero(); qa[1] = v8zero();
        qb[0] = v8zero(); qb[1] = v8zero();
        #pragma unroll 4
        for (int kk = 0; kk < 32; ++kk) {
            int k0 = kk * 4 + 2 * lh;
            v2f kxv = ld2(pga + kk * 4);
            float d0 = xr - gridc(k0);
            float d1 = xr - gridc(k0 + 1);
            v2f aA, aB;
            aA.x = kxv.x * d0 * d0; aA.y = kxv.y * d1 * d1;   // Kx * dx^2
            aB = kxv;                                          // Kx
            #pragma unroll
            for (int t = 0; t < 2; ++t) {
                v2f bv = ld2(pv0 + t * (16 * VPAD) + kk * 4);
                qa[t] = wmma_f32(aA, bv, qa[t]);
                qb[t] = wmma_f32(aB, bv, qb[t]);
            }
        }
        #pragma unroll
        for (int t = 0; t < 2; ++t) {
            int gyt = gyq * 2 + t;
            #pragma unroll
            for (int r = 0; r < 8; ++r) {
                int rn = rbase + ntile * 16 + r + 8 * lh;
                int gy = gyt * 16 + lm;
                float ky = Kyb[rn * OUTN + gy];
                float dyv = s_y[rn] - gridc(gy);
                wdp += s_u[rn] * ky * (qa[t][r] + dyv * dyv * qb[t][r]);
            }
        }
    }

    // ===== beta = REG*log(v+eps);  ot, S=sum(und*beta), sc=sum(und)
    float otp = 0.f, Sp = 0.f, scp = 0.f;
    for (int g = tid; g < GRIDN; g += TPB) {
        int gx = g & 127, gy = g >> 7;
        float vv = s_V[gy * VPAD + gx];
        float beta = REGC * logf(vv + MEPS);
        float ndv = ndb[g], uv = undb[g];
        otp += ndv * beta;
        Sp  += uv * beta;
        scp += uv;
    }
    float sc = blk_reduce(scp, s_red, s_bc, tid);
    float S  = blk_reduce(Sp,  s_red, s_bc, tid);
    float ot = blk_reduce(otp, s_red, s_bc, tid);
    float denom = sc * sc + 1e-8f;

    float lossp = 0.f;
    for (int g = tid; g < GRIDN; g += TPB) {
        int gx = g & 127, gy = g >> 7;
        float vv = s_V[gy * VPAD + gx];
        float beta = REGC * logf(vv + MEPS);
        lossp += undb[g] * (sc / denom * beta - S / denom);
    }
    float loss = blk_reduce(lossp, s_red, s_bc, tid);
    float wd   = blk_reduce(wdp,   s_red, s_bc, tid);

    if (tid == 0) {
        res[b * 3 + 0] = loss;
        res[b * 3 + 1] = wd;
        res[b * 3 + 2] = ot;
    }
}

// ---------------------------------------------------------------------------
// Kernel 3: deterministic final sum over the 8 images
// ---------------------------------------------------------------------------
__global__ void final_reduce(const float* __restrict__ res, float* __restrict__ out) {
    int t = threadIdx.x;
    if (t < 3) {
        float s = 0.f;
        for (int bb = 0; bb < 8; ++bb) s += res[bb * 3 + t];
        out[t] = s;
    }
}

extern "C" void kernel_launch(void* const* d_in, const int* in_sizes, int n_in,
                              void* d_out, int out_size, void* d_ws, size_t ws_size,
                              hipStream_t stream) {
    const float* nd  = (const float*)d_in[0];   // normed_density   8*16384
    const float* und = (const float*)d_in[1];   // unnormed_density 8*16384
    const float* pts = (const float*)d_in[2];   // points           8*1024*2
    // d_in[3] = gt_discrete (unused by reference math)

    float* ws  = (float*)d_ws;
    float* Kx  = ws;                       // 1,048,576 floats (4MB)
    float* Ky  = ws + 1048576;             // 1,048,576 floats (4MB)
    float* res = ws + 2097152;             // 24 floats

    prep_kernel<<<4096, 256, 0, stream>>>(pts, Kx, Ky);
    sinkhorn_kernel<<<8, TPB, 0, stream>>>(nd, und, pts, Kx, Ky, res);
    final_reduce<<<1, 32, 0, stream>>>(res, (float*)d_out);
}